// MPNEncoder_3478923510262
// MI455X (gfx1250) — compile-verified
//
#include <hip/hip_runtime.h>
#include <hip/hip_bf16.h>

typedef float v2f __attribute__((ext_vector_type(2)));
typedef float v8f __attribute__((ext_vector_type(8)));

#define HID 300
#define BM 128
#define BN 64
#define BK 32
#define LDA_PAD (BK + 2)   // 34 floats: even stride -> 8B-aligned float2 reads,
                           // odd/even interleave across lanes -> conflict-free

// ---------------------------------------------------------------------------
// Tiled GEMM on the CDNA5 f32 WMMA path: out[M,300] = epilogue(A[M,K] @ W[K,300])
// MODE 0: A=f_bonds (lda=147);  out0=inp=acc+b,  out1=message=relu(inp)
// MODE 1: A=msg (lda=300);      out0=message=relu(inpbuf + acc + b)
// MODE 2: A=concat(f_atoms[133], a_message[300]) (K=433); out0=relu(acc+b)
// ---------------------------------------------------------------------------
template <int MODE>
__global__ __launch_bounds__(256) void gemm_wmma_f32(
    const float* __restrict__ A, const float* __restrict__ A2,
    const float* __restrict__ W, const float* __restrict__ bias,
    const float* __restrict__ inpbuf, float* __restrict__ out0,
    float* __restrict__ out1, int M, int K) {
  __shared__ __align__(16) float As[BM][LDA_PAD];    // row-major A tile
  __shared__ __align__(16) float WsT[BN][LDA_PAD];   // TRANSPOSED W tile: WsT[col][k]

  const int t    = threadIdx.x;
  const int wave = t >> 5;           // 0..7, one 16-row strip each
  const int lane = t & 31;
  const int half = lane >> 4;        // 0: K={0,1}/M=r ; 1: K={2,3}/M=r+8
  const int lp   = lane & 15;
  const int rowBase = blockIdx.x * BM;
  const int colBase = blockIdx.y * BN;

  v8f acc[4];
  const v8f vzero = {0.f, 0.f, 0.f, 0.f, 0.f, 0.f, 0.f, 0.f};
#pragma unroll
  for (int i = 0; i < 4; ++i) acc[i] = vzero;

  for (int k0 = 0; k0 < K; k0 += BK) {
    // Stage A chunk: 128x32 elements, 16 per thread (coalesced along k)
#pragma unroll
    for (int j = 0; j < 16; ++j) {
      int i = t + 256 * j;
      int r = i >> 5, c = i & 31;
      int grow = rowBase + r, gk = k0 + c;
      float v = 0.f;
      if (grow < M && gk < K) {
        if constexpr (MODE == 2)
          v = (gk < 133) ? A[(size_t)grow * 133 + gk]
                         : A2[(size_t)grow * HID + (gk - 133)];
        else if constexpr (MODE == 0)
          v = A[(size_t)grow * 147 + gk];
        else
          v = A[(size_t)grow * HID + gk];
      }
      As[r][c] = v;
    }
    // Stage W chunk transposed: 32x64 elements, 8 per thread
    // (coalesced read along col; write WsT[col][k], lane-stride 34 -> no conflicts)
#pragma unroll
    for (int j = 0; j < 8; ++j) {
      int i = t + 256 * j;
      int r = i >> 6, c = i & 63;          // r = k local, c = col local
      int gk = k0 + r, gc = colBase + c;
      WsT[c][r] = (gk < K && gc < HID) ? W[(size_t)gk * HID + gc] : 0.f;
    }
    __syncthreads();

#pragma unroll
    for (int kk = 0; kk < BK; kk += 4) {
      // A frag 16x4: contiguous aligned K-pair -> single ds_load_b64
      v2f a = *(const v2f*)&As[wave * 16 + lp][kk + 2 * half];
#pragma unroll
      for (int tt = 0; tt < 4; ++tt) {
        // B frag 4x16: contiguous aligned K-pair from transposed tile
        v2f b = *(const v2f*)&WsT[tt * 16 + lp][kk + 2 * half];
        acc[tt] = __builtin_amdgcn_wmma_f32_16x16x4_f32(
            false, a, false, b, (short)0, acc[tt], false, false);
      }
    }
    __syncthreads();
  }

  // Epilogue: D layout — VGPR r holds M=r (lanes 0-15) / M=r+8 (lanes 16-31)
#pragma unroll
  for (int tt = 0; tt < 4; ++tt) {
    int col = colBase + tt * 16 + lp;
    if (col < HID) {
      float bv = bias[col];
#pragma unroll
      for (int r = 0; r < 8; ++r) {
        int row = rowBase + wave * 16 + r + 8 * half;
        if (row < M) {
          size_t o = (size_t)row * HID + col;
          float v = acc[tt][r] + bv;
          if constexpr (MODE == 0) {
            out0[o] = v;
            out1[o] = fmaxf(v, 0.f);
          } else if constexpr (MODE == 1) {
            out0[o] = fmaxf(inpbuf[o] + v, 0.f);
          } else {
            out0[o] = fmaxf(v, 0.f);
          }
        }
      }
    }
  }
}

// a_message[atom] = sum_{j<6} message[a2b[atom][j]]   (float4-vectorized)
__global__ void atom_gather_sum(const float* __restrict__ message,
                                const int* __restrict__ a2b,
                                float* __restrict__ a_message, int nAtoms) {
  int idx = blockIdx.x * blockDim.x + threadIdx.x;
  if (idx >= nAtoms * 75) return;
  int atom = idx / 75;
  int c = (idx % 75) * 4;
  float4 s = make_float4(0.f, 0.f, 0.f, 0.f);
#pragma unroll
  for (int j = 0; j < 6; ++j) {
    int b = a2b[atom * 6 + j];
    const float4 m = *(const float4*)(message + (size_t)b * HID + c);
    s.x += m.x; s.y += m.y; s.z += m.z; s.w += m.w;
  }
  *(float4*)(a_message + (size_t)atom * HID + c) = s;
}

// msg[bond] = a_message[b2a[bond]] - message[b2revb[bond]]
__global__ void bond_msg(const float* __restrict__ a_message,
                         const float* __restrict__ message,
                         const int* __restrict__ b2a,
                         const int* __restrict__ b2revb,
                         float* __restrict__ msg, int nBonds) {
  int idx = blockIdx.x * blockDim.x + threadIdx.x;
  if (idx >= nBonds * 75) return;
  int bond = idx / 75;
  int c = (idx % 75) * 4;
  const float4 am = *(const float4*)(a_message + (size_t)b2a[bond] * HID + c);
  const float4 rm = *(const float4*)(message + (size_t)b2revb[bond] * HID + c);
  float4 r = make_float4(am.x - rm.x, am.y - rm.y, am.z - rm.z, am.w - rm.w);
  *(float4*)(msg + (size_t)bond * HID + c) = r;
}

__global__ void zero_f32(float* __restrict__ p, int n) {
  int idx = blockIdx.x * blockDim.x + threadIdx.x;
  if (idx < n) p[idx] = 0.f;
}

__global__ void seg_accumulate(const float* __restrict__ hidden,
                               const int* __restrict__ mol_ids,
                               float* __restrict__ sums,
                               float* __restrict__ counts, int nAtoms) {
  int idx = blockIdx.x * blockDim.x + threadIdx.x;
  if (idx >= nAtoms * 75) return;
  int atom = idx / 75;
  int c = (idx % 75) * 4;
  int mol = mol_ids[atom];
  const float4 h = *(const float4*)(hidden + (size_t)atom * HID + c);
  float* s = sums + (size_t)mol * HID + c;
  atomicAdd(s + 0, h.x);
  atomicAdd(s + 1, h.y);
  atomicAdd(s + 2, h.z);
  atomicAdd(s + 3, h.w);
  if ((idx % 75) == 0) atomicAdd(counts + mol, 1.0f);
}

__global__ void seg_finalize(const float* __restrict__ sums,
                             const float* __restrict__ counts,
                             float* __restrict__ out, int nMols) {
  int idx = blockIdx.x * blockDim.x + threadIdx.x;
  if (idx >= nMols * HID) return;
  float c = counts[idx / HID];
  out[idx] = (c > 0.f) ? sums[idx] / c : 0.f;
}

extern "C" void kernel_launch(void* const* d_in, const int* in_sizes, int n_in,
                              void* d_out, int out_size, void* d_ws,
                              size_t ws_size, hipStream_t stream) {
  const float* f_atoms = (const float*)d_in[0];   // [100000,133]
  const float* f_bonds = (const float*)d_in[1];   // [200000,147]
  const int* a2b       = (const int*)d_in[2];     // [100000,6]
  const int* b2a       = (const int*)d_in[3];     // [200000]
  const int* b2revb    = (const int*)d_in[4];     // [200000]
  const int* mol_ids   = (const int*)d_in[5];     // [100000]
  const float* W_i = (const float*)d_in[6];
  const float* b_i = (const float*)d_in[7];
  const float* W_h = (const float*)d_in[8];
  const float* b_h = (const float*)d_in[9];
  const float* W_o = (const float*)d_in[10];
  const float* b_o = (const float*)d_in[11];

  const int nAtoms = 100000, nBonds = 200000, nMols = 4096;

  char* ws = (char*)d_ws;
  float* inp     = (float*)(ws + 0LL);           // 240,000,000 B
  float* message = (float*)(ws + 240000000LL);   // 240,000,000 B
  float* a_msg   = (float*)(ws + 480000000LL);   // 120,000,000 B
  float* msgbuf  = (float*)(ws + 600000000LL);   // 240,000,000 B
  float* hidden  = msgbuf;                       // reuse: msg dead after last W_h GEMM
  float* sums    = (float*)(ws + 840000000LL);   // 4096*300 f32
  float* counts  = sums + (size_t)nMols * HID;   // 4096 f32 (contiguous for zeroing)

  dim3 blk(256);
  dim3 gBond((nBonds + BM - 1) / BM, (HID + BN - 1) / BN);   // 1563 x 5
  dim3 gAtom((nAtoms + BM - 1) / BM, (HID + BN - 1) / BN);   //  782 x 5

  // inp = f_bonds @ W_i + b_i ; message = relu(inp)
  gemm_wmma_f32<0><<<gBond, blk, 0, stream>>>(f_bonds, nullptr, W_i, b_i,
                                              nullptr, inp, message, nBonds, 147);

  for (int d = 0; d < 2; ++d) {  // DEPTH-1 iterations
    atom_gather_sum<<<(nAtoms * 75 + 255) / 256, blk, 0, stream>>>(
        message, a2b, a_msg, nAtoms);
    bond_msg<<<(nBonds * 75 + 255) / 256, blk, 0, stream>>>(
        a_msg, message, b2a, b2revb, msgbuf, nBonds);
    gemm_wmma_f32<1><<<gBond, blk, 0, stream>>>(msgbuf, nullptr, W_h, b_h,
                                                inp, message, nullptr, nBonds, HID);
  }

  // final aggregation + output transform
  atom_gather_sum<<<(nAtoms * 75 + 255) / 256, blk, 0, stream>>>(
      message, a2b, a_msg, nAtoms);
  gemm_wmma_f32<2><<<gAtom, blk, 0, stream>>>(f_atoms, a_msg, W_o, b_o,
                                              nullptr, hidden, nullptr, nAtoms, 433);

  // segment mean readout
  int nz = nMols * HID + nMols;
  zero_f32<<<(nz + 255) / 256, blk, 0, stream>>>(sums, nz);
  seg_accumulate<<<(nAtoms * 75 + 255) / 256, blk, 0, stream>>>(
      hidden, mol_ids, sums, counts, nAtoms);
  seg_finalize<<<(nMols * HID + 255) / 256, blk, 0, stream>>>(
      sums, counts, (float*)d_out, nMols);
}